// DenseGrid_49606872269475
// MI455X (gfx1250) — compile-verified
//
#include <hip/hip_runtime.h>
#include <stdint.h>

#define C_CH 12
#define GD 160
#define GH 160
#define GW 160
#define HW  (GH * GW)            // 25,600
#define DHW (GD * GH * GW)       // 4,096,000

// Transposed (channel-last) layout strides, in floats
#define TX C_CH                  // +1 in x  -> 12
#define TY (GW * C_CH)           // +1 in y  -> 1920
#define TZ (GH * GW * C_CH)      // +1 in z  -> 307200

#define BLK 256

typedef float v4f __attribute__((ext_vector_type(4)));
typedef int   v4i __attribute__((ext_vector_type(4)));

typedef __attribute__((address_space(1))) v4i as1_v4i;
typedef __attribute__((address_space(3))) v4i as3_v4i;

__device__ __forceinline__ void wait_asynccnt0() {
    asm volatile("s_wait_asynccnt 0x0" ::: "memory");
}

// Lower-alignment pair for the fallback path (x0,x1 adjacent in W, only 4B aligned)
struct __attribute__((packed, aligned(4))) fpair { float a, b; };

__device__ __forceinline__ void cell1d(float p, int dim, int& i0, float& fr) {
    float fl = floorf(p);
    int i = (int)fl;
    i = i < 0 ? 0 : i;
    i = i > dim - 2 ? dim - 2 : i;
    i0 = i;
    fr = p - (float)i;           // equals p-floor(p) in-range; (0,1) weights at the top edge
}

// ---------------------------------------------------------------------------
// Trilinear sample from channel-last grid (first in file: snippet shows it).
// The block's 256 xyz coords (3 KB) land in LDS: full blocks via
// GLOBAL_LOAD_ASYNC_TO_LDS_B128 (ASYNCcnt), tail blocks via per-thread copy.
// All threads then read coords from LDS only (ds_load path, no generic
// pointer merge). Each point: 24 contiguous 16B-aligned b128 gather loads
// (4 corner-pair regions x 3 channel blocks), NT output stores.
// ---------------------------------------------------------------------------
__global__ __launch_bounds__(BLK) void sample_cl_kernel(
    const float* __restrict__ xyz,
    const float* __restrict__ gt,
    const float* __restrict__ xyz_min,
    const float* __restrict__ xyz_max,
    float* __restrict__ out, int n_pts)
{
    const uint32_t tid = threadIdx.x;
    const uint32_t blk_base = blockIdx.x * (uint32_t)BLK;
    const uint32_t n = blk_base + tid;

    __shared__ float sxyz[3 * BLK];         // 3 KB of the 320 KB/WGP LDS
    const bool full = (blk_base + BLK) <= (uint32_t)n_pts;   // block-uniform
    if (full) {
        if (tid < (3 * BLK) / 4) {          // 192 lanes x 16B = 3 KB
            const float* src = xyz + (size_t)blk_base * 3 + (size_t)tid * 4;
            __builtin_amdgcn_global_load_async_to_lds_b128(
                (as1_v4i*)src, (as3_v4i*)&sxyz[tid * 4], 0, 0);
        }
        wait_asynccnt0();                   // own wave's ASYNCcnt -> 0
    } else {
        if (n < (uint32_t)n_pts) {          // tail block: per-thread copy
            sxyz[tid * 3 + 0] = __builtin_nontemporal_load(xyz + (size_t)3 * n + 0);
            sxyz[tid * 3 + 1] = __builtin_nontemporal_load(xyz + (size_t)3 * n + 1);
            sxyz[tid * 3 + 2] = __builtin_nontemporal_load(xyz + (size_t)3 * n + 2);
        }
    }
    __syncthreads();                        // all waves' LDS writes visible
    float p0 = sxyz[tid * 3 + 0];           // stride-3 words: conflict-free
    float p1 = sxyz[tid * 3 + 1];
    float p2 = sxyz[tid * 3 + 2];
    if (n >= (uint32_t)n_pts) return;

    float mn0 = xyz_min[0], mn1 = xyz_min[1], mn2 = xyz_min[2];
    float mx0 = xyz_max[0], mx1 = xyz_max[1], mx2 = xyz_max[2];

    // pts.x -> D, pts.y -> H, pts.z -> W (reference's coordinate flip)
    float px = (p2 - mn2) / (mx2 - mn2) * (float)(GW - 1);
    float py = (p1 - mn1) / (mx1 - mn1) * (float)(GH - 1);
    float pz = (p0 - mn0) / (mx0 - mn0) * (float)(GD - 1);

    int x0, y0, z0; float fx, fy, fz;
    cell1d(px, GW, x0, fx);
    cell1d(py, GH, y0, fy);
    cell1d(pz, GD, z0, fz);

    float wx0 = 1.f - fx, wx1 = fx;
    float wy0 = 1.f - fy, wy1 = fy;
    float wz0 = 1.f - fz, wz1 = fz;
    float w0 = wz0 * wy0 * wx0, w1 = wz0 * wy0 * wx1;
    float w2 = wz0 * wy1 * wx0, w3 = wz0 * wy1 * wx1;
    float w4 = wz1 * wy0 * wx0, w5 = wz1 * wy0 * wx1;
    float w6 = wz1 * wy1 * wx0, w7 = wz1 * wy1 * wx1;

    // 32-bit element offset (grid is 49.2M floats < 2^31)
    uint32_t boff = ((uint32_t)z0 * GH + (uint32_t)y0) * (uint32_t)(GW * C_CH)
                  + (uint32_t)x0 * (uint32_t)C_CH;
    const float* base = gt + boff;
    float* op = out + (uint32_t)n * (uint32_t)C_CH;

#pragma unroll
    for (int cb = 0; cb < 3; ++cb) {        // 3 blocks of 4 channels
        const float* b = base + cb * 4;
        v4f a0 = *(const v4f*)(b);                 // z0 y0 x0
        v4f a1 = *(const v4f*)(b + TX);            // z0 y0 x1
        v4f a2 = *(const v4f*)(b + TY);            // z0 y1 x0
        v4f a3 = *(const v4f*)(b + TY + TX);       // z0 y1 x1
        v4f a4 = *(const v4f*)(b + TZ);            // z1 y0 x0
        v4f a5 = *(const v4f*)(b + TZ + TX);       // z1 y0 x1
        v4f a6 = *(const v4f*)(b + TZ + TY);       // z1 y1 x0
        v4f a7 = *(const v4f*)(b + TZ + TY + TX);  // z1 y1 x1
        v4f r;
#pragma unroll
        for (int e = 0; e < 4; ++e) {
            float acc = a0[e] * w0;
            acc = fmaf(a1[e], w1, acc);
            acc = fmaf(a2[e], w2, acc);
            acc = fmaf(a3[e], w3, acc);
            acc = fmaf(a4[e], w4, acc);
            acc = fmaf(a5[e], w5, acc);
            acc = fmaf(a6[e], w6, acc);
            acc = fmaf(a7[e], w7, acc);
            r[e] = acc;
        }
        __builtin_nontemporal_store(r, (v4f*)(op + cb * 4));  // 16B aligned (48B/point)
    }
}

// ---------------------------------------------------------------------------
// Transpose [C, D, H, W] -> [D, H, W, C] (channel-last) into d_ws.
// Each thread handles 4 consecutive x positions: 12 coalesced NT b128 reads,
// 192 contiguous bytes written per lane.
// ---------------------------------------------------------------------------
__global__ __launch_bounds__(BLK) void transpose_kernel(
    const float* __restrict__ g, float* __restrict__ gt)
{
    uint32_t t = blockIdx.x * (uint32_t)BLK + threadIdx.x;
    uint32_t i = t * 4u;                    // element index into D*H*W
    if (i >= (uint32_t)DHW) return;

    v4f v[C_CH];
#pragma unroll
    for (int c = 0; c < C_CH; ++c) {
        uint32_t off = (uint32_t)c * (uint32_t)DHW + i;   // < 2^31/4 elements
        v[c] = __builtin_nontemporal_load((const v4f*)(g + off));
    }

    v4f* o = (v4f*)(gt + i * (uint32_t)C_CH); // 16B aligned: i*12*4 = i*48
#pragma unroll
    for (int xe = 0; xe < 4; ++xe) {        // emit 4 points x 12 channels
        v4f r0, r1, r2;
        r0.x = v[0][xe];  r0.y = v[1][xe];  r0.z = v[2][xe];  r0.w = v[3][xe];
        r1.x = v[4][xe];  r1.y = v[5][xe];  r1.z = v[6][xe];  r1.w = v[7][xe];
        r2.x = v[8][xe];  r2.y = v[9][xe];  r2.z = v[10][xe]; r2.w = v[11][xe];
        o[xe * 3 + 0] = r0;
        o[xe * 3 + 1] = r1;
        o[xe * 3 + 2] = r2;
    }
}

// ---------------------------------------------------------------------------
// Fallback (workspace too small): sample directly from [C,D,H,W] with packed
// 8-byte x-pair loads (48 loads/point instead of 96 scalar loads).
// ---------------------------------------------------------------------------
__global__ __launch_bounds__(BLK) void sample_direct_kernel(
    const float* __restrict__ xyz,
    const float* __restrict__ grid,
    const float* __restrict__ xyz_min,
    const float* __restrict__ xyz_max,
    float* __restrict__ out, int n_pts)
{
    int n = blockIdx.x * BLK + threadIdx.x;
    if (n >= n_pts) return;

    float p0 = __builtin_nontemporal_load(xyz + (size_t)3 * n + 0);
    float p1 = __builtin_nontemporal_load(xyz + (size_t)3 * n + 1);
    float p2 = __builtin_nontemporal_load(xyz + (size_t)3 * n + 2);

    float mn0 = xyz_min[0], mn1 = xyz_min[1], mn2 = xyz_min[2];
    float mx0 = xyz_max[0], mx1 = xyz_max[1], mx2 = xyz_max[2];

    float px = (p2 - mn2) / (mx2 - mn2) * (float)(GW - 1);
    float py = (p1 - mn1) / (mx1 - mn1) * (float)(GH - 1);
    float pz = (p0 - mn0) / (mx0 - mn0) * (float)(GD - 1);

    int x0, y0, z0; float fx, fy, fz;
    cell1d(px, GW, x0, fx);
    cell1d(py, GH, y0, fy);
    cell1d(pz, GD, z0, fz);

    float wx0 = 1.f - fx, wx1 = fx;
    float wy0 = 1.f - fy, wy1 = fy;
    float wz0 = 1.f - fz, wz1 = fz;
    float w0 = wz0 * wy0 * wx0, w1 = wz0 * wy0 * wx1;
    float w2 = wz0 * wy1 * wx0, w3 = wz0 * wy1 * wx1;
    float w4 = wz1 * wy0 * wx0, w5 = wz1 * wy0 * wx1;
    float w6 = wz1 * wy1 * wx0, w7 = wz1 * wy1 * wx1;

    uint32_t boff = (uint32_t)z0 * HW + (uint32_t)y0 * GW + (uint32_t)x0;
    const float* gp = grid + boff;
    float res[C_CH];
#pragma unroll
    for (int c = 0; c < C_CH; ++c) {
        fpair q00 = *(const fpair*)(gp);
        fpair q01 = *(const fpair*)(gp + GW);
        fpair q10 = *(const fpair*)(gp + HW);
        fpair q11 = *(const fpair*)(gp + HW + GW);
        float acc = q00.a * w0;
        acc = fmaf(q00.b, w1, acc);
        acc = fmaf(q01.a, w2, acc);
        acc = fmaf(q01.b, w3, acc);
        acc = fmaf(q10.a, w4, acc);
        acc = fmaf(q10.b, w5, acc);
        acc = fmaf(q11.a, w6, acc);
        acc = fmaf(q11.b, w7, acc);
        res[c] = acc;
        gp += (uint32_t)DHW;
    }

    float* op = out + (size_t)n * C_CH;
    v4f r0 = { res[0], res[1], res[2], res[3] };
    v4f r1 = { res[4], res[5], res[6], res[7] };
    v4f r2 = { res[8], res[9], res[10], res[11] };
    __builtin_nontemporal_store(r0, (v4f*)(op + 0));
    __builtin_nontemporal_store(r1, (v4f*)(op + 4));
    __builtin_nontemporal_store(r2, (v4f*)(op + 8));
}

extern "C" void kernel_launch(void* const* d_in, const int* in_sizes, int n_in,
                              void* d_out, int out_size, void* d_ws, size_t ws_size,
                              hipStream_t stream) {
    const float* xyz     = (const float*)d_in[0];
    const float* grid    = (const float*)d_in[1];
    const float* xyz_min = (const float*)d_in[2];
    const float* xyz_max = (const float*)d_in[3];
    float* out = (float*)d_out;

    int n_pts = in_sizes[0] / 3;
    int blocks = (n_pts + BLK - 1) / BLK;

    size_t need = (size_t)DHW * C_CH * sizeof(float);   // 196,608,000 B
    if (ws_size >= need) {
        float* gt = (float*)d_ws;
        int tblocks = (DHW / 4 + BLK - 1) / BLK;        // 4000
        transpose_kernel<<<tblocks, BLK, 0, stream>>>(grid, gt);
        sample_cl_kernel<<<blocks, BLK, 0, stream>>>(xyz, gt, xyz_min, xyz_max, out, n_pts);
    } else {
        sample_direct_kernel<<<blocks, BLK, 0, stream>>>(xyz, grid, xyz_min, xyz_max, out, n_pts);
    }
}